// FISM_model_24464133718082
// MI455X (gfx1250) — compile-verified
//
#include <hip/hip_runtime.h>
#include <hip/hip_bf16.h>

typedef __attribute__((ext_vector_type(2))) float v2f;
typedef __attribute__((ext_vector_type(4))) float v4f;
typedef __attribute__((ext_vector_type(8))) float v8f;

#define BATCH    4096
#define HIST_LEN 200
#define FACTORS  256
#define NTILES   13   // ceil(200/16)

__global__ __launch_bounds__(256) void fism_kernel(
    const int*   __restrict__ user,
    const int*   __restrict__ item,
    const int*   __restrict__ hist_matrix,
    const float* __restrict__ hist_lens,
    const float* __restrict__ hist_mask,
    const float* __restrict__ Bi,
    const float* __restrict__ Bu,
    const float* __restrict__ Gi,
    const float* __restrict__ Gj,
    float* __restrict__ out_scores,
    float* __restrict__ out_ub,
    float* __restrict__ out_ib,
    float* __restrict__ out_hist,
    float* __restrict__ out_tgt)
{
    __shared__ int   ids_lds[HIST_LEN];
    __shared__ float mask_lds[HIST_LEN];
    __shared__ float tgt_lds[FACTORS];
    __shared__ float acc;

    const int b   = blockIdx.x;
    const int tid = threadIdx.x;
    const int u   = user[b];
    const int it  = item[b];

    // Stage history ids, mask row, and target vector into LDS.
    if (tid < HIST_LEN) {
        ids_lds[tid]  = hist_matrix[(size_t)u * HIST_LEN + tid];
        mask_lds[tid] = hist_mask[(size_t)u * HIST_LEN + tid];
    }
    {   // blockDim.x == FACTORS == 256
        float tv = Gj[(size_t)it * FACTORS + tid];
        tgt_lds[tid] = tv;
        __builtin_nontemporal_store(tv, out_tgt + (size_t)b * FACTORS + tid);
    }
    if (tid == 0) acc = 0.0f;
    __syncthreads();

    const int wave = tid >> 5;
    const int lane = tid & 31;
    const int sel  = lane >> 4;   // 0 -> K 0,1 ; 1 -> K 2,3 (A-matrix layout)
    const int mrow = lane & 15;   // M index within the 16-row tile

    // Lanes 0 and 16 carry column 0 of B; everyone else contributes zeros.
    const float msel = (mrow == 0) ? 1.0f : 0.0f;
    const float* tgtp = tgt_lds + 2 * sel;

    for (int t = wave; t < NTILES; t += 8) {
        const int l0 = t * 16;

        // ---- 1) stream 16 gathered Gi rows to the user_history output ----
        for (int r = 0; r < 16; ++r) {
            const int l = l0 + r;
            if (l < HIST_LEN) {
                const v4f* src = (const v4f*)(Gi + (size_t)ids_lds[l] * FACTORS);
                v4f*       dst = (v4f*)(out_hist + ((size_t)b * HIST_LEN + l) * FACTORS);
                // coalesced 2 x 512B per row; stores are streaming (never re-read)
                __builtin_nontemporal_store(src[lane],      dst + lane);
                __builtin_nontemporal_store(src[lane + 32], dst + lane + 32);
            }
        }

        // ---- 2) per-row dot(target) via V_WMMA_F32_16X16X4_F32 ----
        // A: lane<16 holds (M=lane, K=0,1); lane>=16 holds (M=lane-16, K=2,3)
        // B: only column N=0 non-zero (lane 0: K=0,1 ; lane 16: K=2,3)
        const int   l   = l0 + mrow;
        const int   la  = l < HIST_LEN ? l : HIST_LEN - 1;          // clamp; masked later
        const float* rowp = Gi + (size_t)ids_lds[la] * FACTORS + 2 * sel;

        v8f c = {0.f, 0.f, 0.f, 0.f, 0.f, 0.f, 0.f, 0.f};
        for (int k0 = 0; k0 < FACTORS; k0 += 4) {
            v2f a;
            a.x = rowp[k0];
            a.y = rowp[k0 + 1];
            // Broadcast LDS read on all lanes; zeroed by msel off column 0.
            v2f bb;
            bb.x = msel * tgtp[k0];
            bb.y = msel * tgtp[k0 + 1];
            c = __builtin_amdgcn_wmma_f32_16x16x4_f32(
                    /*neg_a=*/false, a, /*neg_b=*/false, bb,
                    /*c_mod=*/(short)0, c, /*reuse_a=*/false, /*reuse_b=*/false);
        }

        // D[:,0] lives in lane 0 (M=0..7 in c[0..7]) and lane 16 (M=8..15).
        if (mrow == 0) {
            float s = 0.0f;
            #pragma unroll
            for (int j = 0; j < 8; ++j) {
                const int lr = l0 + sel * 8 + j;
                if (lr < HIST_LEN) s += c[j] * mask_lds[lr];
            }
            atomicAdd(&acc, s);                    // LDS float atomic (ds_add)
        }
    }
    __syncthreads();

    if (tid == 0) {
        const float lenf  = hist_lens[u];
        const float coeff = 1.0f / sqrtf(lenf);    // len^(-ALPHA), ALPHA = 0.5
        const float ub    = Bu[u];
        const float ib    = Bi[it];
        const float logit = coeff * acc + ub + ib;
        out_scores[b] = 1.0f / (1.0f + expf(-logit));
        out_ub[b]     = ub;
        out_ib[b]     = ib;
    }
}

extern "C" void kernel_launch(void* const* d_in, const int* in_sizes, int n_in,
                              void* d_out, int out_size, void* d_ws, size_t ws_size,
                              hipStream_t stream) {
    const int*   user        = (const int*)d_in[0];
    const int*   item        = (const int*)d_in[1];
    const int*   hist_matrix = (const int*)d_in[2];
    const float* hist_lens   = (const float*)d_in[3];
    const float* hist_mask   = (const float*)d_in[4];
    const float* Bi          = (const float*)d_in[5];
    const float* Bu          = (const float*)d_in[6];
    const float* Gi          = (const float*)d_in[7];
    const float* Gj          = (const float*)d_in[8];

    // Outputs concatenated flat in reference return order:
    // scores[B], user_bias[B], item_bias[B], user_history[B*L*F], target[B*F]
    float* out        = (float*)d_out;
    float* out_scores = out;
    float* out_ub     = out + BATCH;
    float* out_ib     = out + 2 * (size_t)BATCH;
    float* out_hist   = out + 3 * (size_t)BATCH;
    float* out_tgt    = out_hist + (size_t)BATCH * HIST_LEN * FACTORS;

    fism_kernel<<<BATCH, 256, 0, stream>>>(user, item, hist_matrix, hist_lens,
                                           hist_mask, Bi, Bu, Gi, Gj,
                                           out_scores, out_ub, out_ib,
                                           out_hist, out_tgt);
}